// HNetBoundaryPredictor_88648124991228
// MI455X (gfx1250) — compile-verified
//
#include <hip/hip_runtime.h>

// ---------------------------------------------------------------------------
// HNet boundary predictor, MI455X (gfx1250, wave32, WMMA + async-to-LDS)
//   B=16, L=1500, D=1280
//   q = X[:, :-1] @ Wq^T ; k = X[:, 1:] @ Wk^T ; cos = <q^,k^> ;
//   p = clip((1-cos)/2,0,1), pad p[0]=1 ; mask = p>0.5 ; stable compaction.
//
// f32 WMMA (16x16x4) is mandatory: mask is a sign test on cos ~ N(0,1/1280);
// bf16-family inputs leave ~2e-5 cos error -> flipped masks -> compaction
// diverges. f32 accumulate keeps error ~1e-7.
//
// A = W rows (e x d), B = X^T (d x token) => C tile = Q^T; per-token norms
// and q.k reduce to per-lane scalars + one shfl_xor(16).
//
// M-tile = 64 pairs/WG (4 row-blocks share every W fragment). Staging uses
// double-buffered GLOBAL_LOAD_ASYNC_TO_LDS_B128 (ASYNCcnt): exactly 13 async
// instructions per wave per chunk (branchless clamped tails keep per-wave
// counts uniform), so "s_wait_asynccnt 13" after issuing the next chunk
// retires the previous one (async loads complete in order).
//
// Round structure is an explicit loop nest so accumulator zeroing and the
// e-tile fold are unconditional at round scope (round 2 showed the compiler
// lowering "if (c==0) cq={}" as 64 v_cndmask selects per chunk iteration).
// ---------------------------------------------------------------------------

#define BATCH 16
#define SEQL  1500
#define DIM   1280
#define PAIRS (SEQL - 1)          // 1499 adjacent token pairs
#define EPSN  1e-12f

#define MROWS 64                  // token pairs per WG (4 x 16-row blocks)
#define XROWS (MROWS + 1)         // 65 staged tokens (k is shifted by one)
#define KC    32                  // K-chunk (floats) staged per buffer
#define LSTR  36                  // padded LDS row stride -> conflict-free b64
#define WAVES 4
#define NCH   (DIM / KC)          // 40 K-chunks per e-round
#define ROUNDS (DIM / 16 / WAVES) // 20 e-rounds (each wave: one 16-e tile)

typedef __attribute__((ext_vector_type(2))) float v2f;
typedef __attribute__((ext_vector_type(8))) float v8f;

__device__ __forceinline__ void async_b128(unsigned ldsoff, unsigned goff,
                                           const float* base) {
    // GVS mode: mem = SADDR(64) + VADDR(32-bit byte offset); VDST = LDS addr
    asm volatile("global_load_async_to_lds_b128 %0, %1, %2"
                 :: "v"(ldsoff), "v"(goff), "s"(base) : "memory");
}

// ---------------------------------------------------------------------------
// Kernel 1: fused WMMA GEMM + cosine + boundary probability outputs
// grid: (ceil(PAIRS/64), BATCH), block: 128 (4 waves)
// ---------------------------------------------------------------------------
__global__ __launch_bounds__(128)
void hnet_boundary_gemm(const float* __restrict__ X,   // (B, L, D)
                        const float* __restrict__ Wq,  // (D, D) row-major
                        const float* __restrict__ Wk,  // (D, D)
                        float* __restrict__ out_bpb,   // (B, L, 2)
                        float* __restrict__ out_mask,  // (B, L)
                        float* __restrict__ out_sel,   // (B, L)
                        int*   __restrict__ ws_mask)   // (B, L) int
{
    const int b    = blockIdx.y;
    const int l0   = blockIdx.x * MROWS;    // base pair index
    const int tid  = threadIdx.x;
    const int wave = tid >> 5;
    const int lane = tid & 31;
    const int hv   = lane >> 4;             // half-wave: K sub-pair select
    const int l16  = lane & 15;             // A: row e ; B: column token

    __shared__ float sW[2][2][64 * LSTR];   // [buf][mat] 64 W rows x chunk
    __shared__ float sX[2][XROWS * LSTR];   // [buf] tokens l0 .. l0+64
    __shared__ float sRed[3 * WAVES * MROWS];

    const float* Xb = X + (size_t)b * SEQL * DIM;

    // --- async staging of one K-chunk: exactly 13 async instrs per wave ---
    auto stage = [&](int wrow0, int kc, int buf) {
        #pragma unroll
        for (int it = 0; it < 4; ++it) {    // Wq: 64 rows x 8 float4
            const int e  = tid + it * 128;  // [0, 512)
            const int r  = e >> 3;
            const int c4 = (e & 7) << 2;
            const unsigned goff = (unsigned)(((wrow0 + r) * DIM + kc + c4) * 4);
            async_b128((unsigned)(size_t)&sW[buf][0][r * LSTR + c4], goff, Wq);
        }
        #pragma unroll
        for (int it = 0; it < 4; ++it) {    // Wk
            const int e  = tid + it * 128;
            const int r  = e >> 3;
            const int c4 = (e & 7) << 2;
            const unsigned goff = (unsigned)(((wrow0 + r) * DIM + kc + c4) * 4);
            async_b128((unsigned)(size_t)&sW[buf][1][r * LSTR + c4], goff, Wk);
        }
        #pragma unroll
        for (int it = 0; it < 5; ++it) {    // X: 65 rows x 8 float4 (=520) in
            const int e = tid + it * 128;   // 640 slots; tail lanes clamp to
            int r = e >> 3;                 // row 64 -> duplicate identical
            if (r > MROWS) r = MROWS;       // writes (branchless, uniform)
            const int c4 = (e & 7) << 2;
            int tok = l0 + r; if (tok > SEQL - 1) tok = SEQL - 1;
            const unsigned goff = (unsigned)((tok * DIM + kc + c4) * 4);
            async_b128((unsigned)(size_t)&sX[buf][r * LSTR + c4], goff, Xb);
        }
    };

    float sq[4]  = {0.f, 0.f, 0.f, 0.f};
    float sk[4]  = {0.f, 0.f, 0.f, 0.f};
    float sqk[4] = {0.f, 0.f, 0.f, 0.f};

    stage(0, 0, 0);                         // prologue: chunk 0 in flight
    int buf = 0;

    for (int round = 0; round < ROUNDS; ++round) {
        v8f cq[4], ck[4];                   // unconditional reset, round scope
        #pragma unroll
        for (int rb = 0; rb < 4; ++rb) { cq[rb] = {}; ck[rb] = {}; }

        for (int c = 0; c < NCH; ++c) {
            // next chunk coordinates (incremental, no division)
            const int lastc  = (c == NCH - 1) ? 1 : 0;
            const int nround = round + lastc;
            const int nkc    = lastc ? 0 : (c + 1) * KC;
            const int nbuf   = buf ^ 1;

            if (nround < ROUNDS) {          // overlap: issue next chunk, then
                stage(nround * MROWS, nkc, nbuf);
                asm volatile("s_wait_asynccnt 13" ::: "memory");
            } else {
                asm volatile("s_wait_asynccnt 0" ::: "memory");
            }
            __syncthreads();                // chunk visible to all waves

            // ---- WMMA over this chunk: W frag shared by 4 row-blocks ----
            {
                const float* wqp = &sW[buf][0][(wave * 16 + l16) * LSTR + 2 * hv];
                const float* wkp = &sW[buf][1][(wave * 16 + l16) * LSTR + 2 * hv];
                const float* xp  = &sX[buf][l16 * LSTR + 2 * hv];
                #pragma unroll
                for (int k0 = 0; k0 < KC; k0 += 4) {
                    v2f aq = *(const v2f*)(wqp + k0);
                    v2f ak = *(const v2f*)(wkp + k0);
                    #pragma unroll
                    for (int rb = 0; rb < 4; ++rb) {
                        v2f bq = *(const v2f*)(xp + (rb * 16    ) * LSTR + k0);
                        v2f bk = *(const v2f*)(xp + (rb * 16 + 1) * LSTR + k0);
                        cq[rb] = __builtin_amdgcn_wmma_f32_16x16x4_f32(
                                    false, aq, false, bq, (short)0, cq[rb], false, false);
                        ck[rb] = __builtin_amdgcn_wmma_f32_16x16x4_f32(
                                    false, ak, false, bk, (short)0, ck[rb], false, false);
                    }
                }
            }
            __syncthreads();                // done reading buf before reissue
            buf = nbuf;
        }

        // ---- end of e-round: fold this e-tile (unconditional) ----
        #pragma unroll
        for (int rb = 0; rb < 4; ++rb) {
            #pragma unroll
            for (int j = 0; j < 8; ++j) {
                sq[rb]  += cq[rb][j] * cq[rb][j];
                sk[rb]  += ck[rb][j] * ck[rb][j];
                sqk[rb] += cq[rb][j] * ck[rb][j];
            }
        }
    }

    // token t of block rb lives in lanes t and t+16 (C cols = lane%16)
    #pragma unroll
    for (int rb = 0; rb < 4; ++rb) {
        sq[rb]  += __shfl_xor(sq[rb],  16, 32);
        sk[rb]  += __shfl_xor(sk[rb],  16, 32);
        sqk[rb] += __shfl_xor(sqk[rb], 16, 32);
    }
    __syncthreads();
    if (hv == 0) {
        #pragma unroll
        for (int rb = 0; rb < 4; ++rb) {
            const int t = rb * 16 + l16;
            sRed[(0 * WAVES + wave) * MROWS + t] = sq[rb];
            sRed[(1 * WAVES + wave) * MROWS + t] = sk[rb];
            sRed[(2 * WAVES + wave) * MROWS + t] = sqk[rb];
        }
    }
    __syncthreads();

    if (tid < MROWS) {
        float tq = 0.f, tk = 0.f, tqk = 0.f;
        #pragma unroll
        for (int w = 0; w < WAVES; ++w) {
            tq  += sRed[(0 * WAVES + w) * MROWS + tid];
            tk  += sRed[(1 * WAVES + w) * MROWS + tid];
            tqk += sRed[(2 * WAVES + w) * MROWS + tid];
        }
        const int j = l0 + tid;                       // pair (x_j, x_{j+1})
        if (j <= SEQL - 2) {
            const float nq = fmaxf(sqrtf(tq), EPSN);
            const float nk = fmaxf(sqrtf(tk), EPSN);
            const float cosv = tqk / (nq * nk);
            float p = 0.5f * (1.f - cosv);
            p = fminf(fmaxf(p, 0.f), 1.f);
            const int i = j + 1;                      // prob index (pad at 0)
            const size_t o = (size_t)b * SEQL + i;
            out_bpb[o * 2 + 0] = 1.f - p;
            out_bpb[o * 2 + 1] = p;
            const int m = (p > 0.5f) ? 1 : 0;         // argmax first-max tie->0
            out_mask[o] = (float)m;
            out_sel[o]  = fmaxf(p, 1.f - p);
            ws_mask[o]  = m;
        }
        if (blockIdx.x == 0 && tid == 0) {            // padded position l=0
            const size_t o = (size_t)b * SEQL;
            out_bpb[o * 2 + 0] = 0.f;
            out_bpb[o * 2 + 1] = 1.f;
            out_mask[o] = 1.f;
            out_sel[o]  = 1.f;
            ws_mask[o]  = 1;
        }
    }
}

// ---------------------------------------------------------------------------
// Kernel 2: per-batch stable compaction scan (1500 elems, 1024 threads x 2)
// ---------------------------------------------------------------------------
__global__ __launch_bounds__(1024)
void hnet_boundary_scan(const int* __restrict__ ws_mask,
                        int* __restrict__ src_idx,   // (B, L)
                        int* __restrict__ nb)        // (B,)
{
    const int b = blockIdx.x;
    const int t = threadIdx.x;
    __shared__ int sSum[1024];

    const int l0 = 2 * t, l1 = 2 * t + 1;
    int m0 = (l0 < SEQL) ? ws_mask[b * SEQL + l0] : 0;
    int m1 = (l1 < SEQL) ? ws_mask[b * SEQL + l1] : 0;
    const int s = m0 + m1;
    sSum[t] = s;
    __syncthreads();
    for (int off = 1; off < 1024; off <<= 1) {   // Hillis-Steele inclusive
        int v = 0;
        if (t >= off) v = sSum[t - off];
        __syncthreads();
        sSum[t] += v;
        __syncthreads();
    }
    const int incl = sSum[t];
    const int excl = incl - s;
    if (m0) src_idx[b * SEQL + excl]      = l0;
    if (m1) src_idx[b * SEQL + excl + m0] = l1;
    if (t == 1023) nb[b] = incl;
}

// ---------------------------------------------------------------------------
// Kernel 3: gather boundary rows to front, zero-fill the rest
// grid: (L, B), block: 256
// ---------------------------------------------------------------------------
__global__ __launch_bounds__(256)
void hnet_boundary_gather(const float* __restrict__ X,
                          const int* __restrict__ src_idx,
                          const int* __restrict__ nb,
                          float* __restrict__ out_comp)  // (B, L, D)
{
    const int b = blockIdx.y;
    const int i = blockIdx.x;
    const int t = threadIdx.x;
    float4* d = (float4*)(out_comp + ((size_t)b * SEQL + i) * DIM);
    if (i < nb[b]) {
        const int src = src_idx[b * SEQL + i];
        const float4* s = (const float4*)(X + ((size_t)b * SEQL + src) * DIM);
        for (int c = t; c < DIM / 4; c += 256) d[c] = s[c];
    } else {
        const float4 z = {0.f, 0.f, 0.f, 0.f};
        for (int c = t; c < DIM / 4; c += 256) d[c] = z;
    }
}

// ---------------------------------------------------------------------------
extern "C" void kernel_launch(void* const* d_in, const int* in_sizes, int n_in,
                              void* d_out, int out_size, void* d_ws, size_t ws_size,
                              hipStream_t stream) {
    const float* X  = (const float*)d_in[0];   // (16,1500,1280) f32
    const float* Wq = (const float*)d_in[1];   // (1280,1280)    f32
    const float* Wk = (const float*)d_in[2];   // (1280,1280)    f32

    float* out = (float*)d_out;
    float* out_bpb  = out;                                   // (B,L,2)
    float* out_mask = out + (size_t)BATCH * SEQL * 2;        // (B,L)
    float* out_sel  = out_mask + (size_t)BATCH * SEQL;       // (B,L,1)
    float* out_comp = out_sel + (size_t)BATCH * SEQL;        // (B,L,D)

    int* ws_mask = (int*)d_ws;                               // (B,L)
    int* src_idx = ws_mask + BATCH * SEQL;                   // (B,L)
    int* nb      = src_idx + BATCH * SEQL;                   // (B,)

    dim3 g1((PAIRS + MROWS - 1) / MROWS, BATCH);
    hnet_boundary_gemm<<<g1, 128, 0, stream>>>(X, Wq, Wk, out_bpb, out_mask,
                                               out_sel, ws_mask);
    hnet_boundary_scan<<<BATCH, 1024, 0, stream>>>(ws_mask, src_idx, nb);
    dim3 g3(SEQL, BATCH);
    hnet_boundary_gather<<<g3, 256, 0, stream>>>(X, src_idx, nb, out_comp);
}